// GAT_module_31619549233984
// MI455X (gfx1250) — compile-verified
//
#include <hip/hip_runtime.h>
#include <math.h>

// Problem dims
#define N_  64
#define C_  64
#define T_  256
#define V_  25
#define H_  8
#define O_  64

// Main-kernel tiling
#define TB   16            // timesteps per block
#define JT   (TB * V_)     // 400: flattened (t,v) columns, exactly 25 WMMA N-tiles
#define VSTR 404           // padded LDS row stride for vv (bank-conflict free)
#define WSTR 72            // padded LDS row stride for Wv^T (bank-conflict free)

typedef float v2f __attribute__((ext_vector_type(2)));
typedef float v8f __attribute__((ext_vector_type(8)));

// ---- optional CDNA5 async global->LDS path ----
#define HAS_ASYNC 0
#if defined(__has_builtin)
#if __has_builtin(__builtin_amdgcn_global_load_async_to_lds_b128)
#undef HAS_ASYNC
#define HAS_ASYNC 1
#endif
#endif

#if HAS_ASYNC
// builtin prototype (from clang diagnostic): (v4i AS1*, v4i AS3*, Imm int, Imm int)
typedef int v4i __attribute__((vector_size(16)));
typedef __attribute__((address_space(1))) v4i g_v4i;
typedef __attribute__((address_space(3))) v4i l_v4i;
#endif

static __device__ inline void wait_async_done() {
#if HAS_ASYNC
#if __has_builtin(__builtin_amdgcn_s_wait_asynccnt)
    __builtin_amdgcn_s_wait_asynccnt(0);
#else
    asm volatile("s_wait_asynccnt 0" ::: "memory");
#endif
#endif
}

// ------------------------------------------------------------------
// Kernel 1: temporal mean  x(N,C,T,V) -> xmean(N,C,V)
// one wave per (n,c); lane v sums over T (row-contiguous 100B reads/iter)
// ------------------------------------------------------------------
__global__ __launch_bounds__(32) void mean_kernel(const float* __restrict__ x,
                                                  float* __restrict__ xmean) {
    int nc   = blockIdx.x;          // n*C + c
    int lane = threadIdx.x;
    if (lane >= V_) return;
    const float* p = x + (size_t)nc * (T_ * V_) + lane;
    float s = 0.f;
    #pragma unroll 8
    for (int t = 0; t < T_; ++t) s += p[t * V_];
    xmean[nc * V_ + lane] = s * (1.0f / T_);
}

// ------------------------------------------------------------------
// Kernel 2: q,k, tanh attention, head reprojection, mask
//   masked(n,o,u,v) = alpha*(sum_h Wr[o,h]*tanh(q[n,h,u]-k[n,h,v]) + br[o]) + A[u,v]
// one block per n (all tiny; stays in 28KB LDS)
// ------------------------------------------------------------------
__global__ __launch_bounds__(256) void masked_kernel(
    const float* __restrict__ xmean, const float* __restrict__ A,
    const float* __restrict__ alpha_p,
    const float* __restrict__ Wq, const float* __restrict__ bq,
    const float* __restrict__ Wk, const float* __restrict__ bk,
    const float* __restrict__ Wr, const float* __restrict__ br,
    float* __restrict__ maskedws) {
    __shared__ float xm[C_ * V_];        // 1600
    __shared__ float qs[H_ * V_];        // 200
    __shared__ float ks[H_ * V_];        // 200
    __shared__ float at[H_ * V_ * V_];   // 5000

    int n = blockIdx.x, tid = threadIdx.x;
    for (int i = tid; i < C_ * V_; i += 256) xm[i] = xmean[(size_t)n * C_ * V_ + i];
    __syncthreads();

    for (int i = tid; i < H_ * V_; i += 256) {
        int h = i / V_, v = i % V_;
        float aq = bq[h], ak = bk[h];
        for (int c = 0; c < C_; ++c) {
            float xv = xm[c * V_ + v];
            aq += Wq[h * C_ + c] * xv;
            ak += Wk[h * C_ + c] * xv;
        }
        qs[i] = aq; ks[i] = ak;
    }
    __syncthreads();

    for (int i = tid; i < H_ * V_ * V_; i += 256) {
        int h = i / (V_ * V_), r = i % (V_ * V_);
        int u = r / V_, v = r % V_;
        at[i] = tanhf(qs[h * V_ + u] - ks[h * V_ + v]);
    }
    __syncthreads();

    float alpha = alpha_p[0];
    for (int i = tid; i < O_ * V_ * V_; i += 256) {
        int o = i / (V_ * V_), r = i % (V_ * V_);
        float acc = br[o];
        #pragma unroll
        for (int h = 0; h < H_; ++h) acc += Wr[o * H_ + h] * at[h * V_ * V_ + r];
        maskedws[(size_t)n * O_ * V_ * V_ + i] = alpha * acc + A[r];
    }
}

// ------------------------------------------------------------------
// Kernel 3 (the heavy one): per (n, 16-t tile)
//   LDS-staged:  xs = x[n,:,t0:t0+16,:]  (100KB, async-to-LDS)
//   GEMM1 (WMMA f32 16x16x4): vv[o, t*25+v] = Wv[o,:]·xs[:,j] + bv[o]  -> LDS (101KB)
//   GEMM2 (WMMA f32 16x16x4): out[n,o,t0+t,u] = sum_v vv[o,t,v]*masked[n,o,u,v]
// 256 threads = 8 waves; wave-uniform tile distribution keeps EXEC all-1s at WMMAs.
// ------------------------------------------------------------------
__global__ __launch_bounds__(256) void fused_kernel(
    const float* __restrict__ x, const float* __restrict__ Wv,
    const float* __restrict__ bv, const float* __restrict__ maskedws,
    float* __restrict__ out) {
    __shared__ float xs[C_ * JT];      // 102,400 B
    __shared__ float vvs[O_ * VSTR];   // 103,424 B
    __shared__ float wvsT[C_ * WSTR];  //  18,432 B   (total ~224 KB of the 320 KB WGP LDS)

    int n = blockIdx.x;
    int t0 = blockIdx.y * TB;
    int tid  = threadIdx.x;
    int wave = tid >> 5;
    int lane = tid & 31;
    int lr   = lane & 15;
    int hi   = lane >> 4;              // 0: K rows kk,kk+1 / M rows r ; 1: kk+2,kk+3 / r+8

    // ---- stage x tile: 64 rows of 400 contiguous floats each ----
    const float* xbase = x + (size_t)n * C_ * (T_ * V_) + (size_t)t0 * V_;
#if HAS_ASYNC
    for (int i = tid; i < C_ * (JT / 4); i += 256) {
        int row = i / (JT / 4);
        int c4  = (i % (JT / 4)) * 4;
        __builtin_amdgcn_global_load_async_to_lds_b128(
            (g_v4i*)(xbase + (size_t)row * (T_ * V_) + c4),
            (l_v4i*)(xs + row * JT + c4), 0, 0);
    }
#else
    for (int i = tid; i < C_ * (JT / 4); i += 256) {
        int row = i / (JT / 4);
        int c4  = (i % (JT / 4)) * 4;
        *(float4*)(xs + row * JT + c4) =
            *(const float4*)(xbase + (size_t)row * (T_ * V_) + c4);
    }
#endif
    // Wv transposed into LDS: wvsT[c][o]
    for (int i = tid; i < O_ * C_; i += 256) {
        int o = i / C_, c = i % C_;
        wvsT[c * WSTR + o] = Wv[i];
    }
    wait_async_done();
    __syncthreads();

    // ---- GEMM1: M=O (4 tiles) x N=JT (25 tiles) x K=C (16 steps of 4) ----
    for (int tt = wave; tt < 4 * 25; tt += 8) {
        int m0 = (tt / 25) * 16;
        int j0 = (tt % 25) * 16;
        v8f acc;
        #pragma unroll
        for (int r = 0; r < 8; ++r) acc[r] = bv[m0 + r + 8 * hi];
        #pragma unroll
        for (int kk = 0; kk < C_; kk += 4) {
            int ka = kk + 2 * hi;
            v2f a, b;
            a.x = wvsT[ka * WSTR + m0 + lr];
            a.y = wvsT[(ka + 1) * WSTR + m0 + lr];
            b.x = xs[ka * JT + j0 + lr];
            b.y = xs[(ka + 1) * JT + j0 + lr];
            acc = __builtin_amdgcn_wmma_f32_16x16x4_f32(
                false, a, false, b, (short)0, acc, false, false);
        }
        #pragma unroll
        for (int r = 0; r < 8; ++r)
            vvs[(m0 + r + 8 * hi) * VSTR + j0 + lr] = acc[r];
    }
    __syncthreads();

    // ---- GEMM2: per o: M=16 (t) x N=25 (u, 2 tiles) x K=25 (v, pad->28) ----
    const float* mb = maskedws + (size_t)n * O_ * V_ * V_;
    for (int tt = wave; tt < O_ * 2; tt += 8) {
        int o  = tt >> 1;
        int u0 = (tt & 1) * 16;
        const float* mo = mb + o * V_ * V_;
        v8f acc;
        #pragma unroll
        for (int r = 0; r < 8; ++r) acc[r] = 0.f;
        #pragma unroll
        for (int kk = 0; kk < 28; kk += 4) {
            int kv = kk + 2 * hi;
            int u  = u0 + lr;
            v2f a, b;
            a.x = (kv     < V_) ? vvs[o * VSTR + lr * V_ + kv]     : 0.f;
            a.y = (kv + 1 < V_) ? vvs[o * VSTR + lr * V_ + kv + 1] : 0.f;
            b.x = (u < V_ && kv     < V_) ? mo[u * V_ + kv]     : 0.f;
            b.y = (u < V_ && kv + 1 < V_) ? mo[u * V_ + kv + 1] : 0.f;
            acc = __builtin_amdgcn_wmma_f32_16x16x4_f32(
                false, a, false, b, (short)0, acc, false, false);
        }
        int u = u0 + lr;
        if (u < V_) {
            #pragma unroll
            for (int r = 0; r < 8; ++r) {
                int t = t0 + r + 8 * hi;
                out[(((size_t)n * O_ + o) * T_ + t) * V_ + u] = acc[r];
            }
        }
    }
}

// ------------------------------------------------------------------
extern "C" void kernel_launch(void* const* d_in, const int* in_sizes, int n_in,
                              void* d_out, int out_size, void* d_ws, size_t ws_size,
                              hipStream_t stream) {
    const float* x     = (const float*)d_in[0];
    const float* A     = (const float*)d_in[1];
    const float* alpha = (const float*)d_in[2];
    const float* Wq    = (const float*)d_in[3];
    const float* bq    = (const float*)d_in[4];
    const float* Wk    = (const float*)d_in[5];
    const float* bk    = (const float*)d_in[6];
    const float* Wv    = (const float*)d_in[7];
    const float* bv    = (const float*)d_in[8];
    const float* Wr    = (const float*)d_in[9];
    const float* br    = (const float*)d_in[10];
    float* out = (float*)d_out;

    float* ws     = (float*)d_ws;
    float* xmean  = ws;                              // N*C*V   = 102,400 f32
    float* masked = ws + (size_t)N_ * C_ * V_;       // N*O*V*V = 2,560,000 f32 (~10.7MB total)

    mean_kernel<<<dim3(N_ * C_), dim3(32), 0, stream>>>(x, xmean);
    masked_kernel<<<dim3(N_), dim3(256), 0, stream>>>(xmean, A, alpha, Wq, bq,
                                                      Wk, bk, Wr, br, masked);
    fused_kernel<<<dim3(N_, T_ / TB), dim3(256), 0, stream>>>(x, Wv, bv, masked, out);
}